// SLIM_65360812310621
// MI455X (gfx1250) — compile-verified
//
#include <hip/hip_runtime.h>
#include <hip/hip_bf16.h>

typedef __attribute__((ext_vector_type(2))) float v2f;
typedef __attribute__((ext_vector_type(8))) float v8f;

#define K_DIM 64

__global__ void slim_zero_out_kernel(float* __restrict__ out, int n) {
    int i = blockIdx.x * blockDim.x + threadIdx.x;
    if (i < n) out[i] = 0.0f;
}

// One wave32 processes 16 edges of the COO matrix.
// contrib[16x64] = diag(vals) @ G, G[j,:] = W[cols[j],:], computed with
// V_WMMA_F32_16X16X4_F32 (4 column tiles x 4 inner chunks), then
// scatter-added into out[rows[j]*64 + k] with global f32 atomics.
__global__ __launch_bounds__(256)
void slim_coo_spmm_wmma_kernel(const float* __restrict__ vals,
                               const float* __restrict__ W,
                               const int* __restrict__ rows,
                               const int* __restrict__ cols,
                               float* __restrict__ out,
                               int nnz) {
    const int lane = threadIdx.x & 31;
    const int wave = (blockIdx.x * (blockDim.x >> 5)) + (threadIdx.x >> 5);
    const int base = wave * 16;
    const int m    = lane & 15;      // edge-in-batch (for A/rows) or column-in-tile (for B/D)
    const bool hi  = lane >= 16;

    // Coalesced batch loads; tail edges get v=0 so they contribute exactly 0.
    int  e     = base + m;
    bool valid = (e < nnz);
    int  ec    = valid ? e : 0;
    float v_m  = valid ? vals[ec] : 0.0f;
    int   r_m  = rows[ec];
    int   c_m  = cols[ec];

    // A = diag(v) split into 16x4 chunks. ISA layout: lane<16 -> (M=m, K=0/1),
    // lane>=16 -> (M=m, K=2/3) across the 2 VGPRs.
    v2f a[4];
#pragma unroll
    for (int c = 0; c < 4; ++c) {
        int k0 = 4 * c + (hi ? 2 : 0);
        a[c].x = (m == k0)     ? v_m : 0.0f;
        a[c].y = (m == k0 + 1) ? v_m : 0.0f;
    }

#pragma unroll
    for (int n = 0; n < 4; ++n) {            // 4 tiles of 16 K-columns
        v8f acc = {};
#pragma unroll
        for (int c = 0; c < 4; ++c) {        // inner dim (edges) in chunks of 4
            int e0   = 4 * c + (hi ? 2 : 0); // B rows held by this half-wave
            int col0 = __shfl(c_m, e0, 32);
            int col1 = __shfl(c_m, e0 + 1, 32);
            v2f b;
            // half-wave coalesced 64B gathers from L2-resident W
            b.x = W[col0 * K_DIM + n * 16 + m];
            b.y = W[col1 * K_DIM + n * 16 + m];
            acc = __builtin_amdgcn_wmma_f32_16x16x4_f32(
                /*neg_a=*/false, a[c], /*neg_b=*/false, b,
                /*c_mod=*/(short)0, acc, /*reuse_a=*/false, /*reuse_b=*/false);
        }
        // D layout: lane<16 -> N=m, VGPR r -> M=r ; lane>=16 -> M=r+8.
#pragma unroll
        for (int r = 0; r < 8; ++r) {
            int mrow = r + (hi ? 8 : 0);          // edge index in batch
            int urow = __shfl(r_m, mrow, 32);     // its user row
            atomicAdd(&out[urow * K_DIM + n * 16 + m], acc[r]);
        }
    }
}

extern "C" void kernel_launch(void* const* d_in, const int* in_sizes, int n_in,
                              void* d_out, int out_size, void* d_ws, size_t ws_size,
                              hipStream_t stream) {
    const float* vals = (const float*)d_in[0];   // feedback_vals  [NNZ]
    const float* W    = (const float*)d_in[1];   // dense_weight_slice [N_ITEMS, 64]
    const int*   rows = (const int*)  d_in[2];   // feedback_rows  [NNZ]
    const int*   cols = (const int*)  d_in[3];   // feedback_cols  [NNZ]
    float* out = (float*)d_out;                  // [N_USERS, 64]
    int nnz = in_sizes[0];

    // Zero the (poisoned) output accumulator first; same stream => ordered.
    int zt = 256;
    int zb = (out_size + zt - 1) / zt;
    slim_zero_out_kernel<<<zb, zt, 0, stream>>>(out, out_size);

    // One wave32 per 16 edges; 256 threads = 8 waves per block.
    int waves  = (nnz + 15) / 16;
    int blocks = (waves + 7) / 8;
    slim_coo_spmm_wmma_kernel<<<blocks, 256, 0, stream>>>(vals, W, rows, cols, out, nnz);
}